// MMOE_48412871360789
// MI455X (gfx1250) — compile-verified
//
#include <hip/hip_runtime.h>
#include <hip/hip_bf16.h>

#define Vv 100000
#define En 8
#define Hn 64
#define On 32
#define THn 32
#define Tn 2
#define Bn 16384
#define NNZn 819200

typedef __attribute__((ext_vector_type(2))) float v2f;
typedef __attribute__((ext_vector_type(8))) float v8f;

__device__ __forceinline__ int lower_bound(const int* __restrict__ idx, int n, int v) {
    int lo = 0, hi = n;
    while (lo < hi) {
        int mid = (lo + hi) >> 1;
        if (idx[mid] < v) lo = mid + 1; else hi = mid;
    }
    return lo;
}

// Kernel 1: one wave per batch row. Single sweep over that row's nnz range:
// accumulate all 8 experts' pooled H=64 vector (2 dims/lane, float2 coalesced)
// and both tasks' 8 gate logits (lanes 0..15). Emits relu(pool + bias).
__global__ void mmoe_pool_kernel(const int* __restrict__ index,
                                 const int* __restrict__ feats,
                                 const float* __restrict__ values,
                                 const float* __restrict__ w_exp0,
                                 const float* __restrict__ b_exp0,
                                 const float* __restrict__ w_gates,
                                 float* __restrict__ h_out,     // (E,B,H)
                                 float* __restrict__ logits)    // (T,B,E)
{
    const int wave = threadIdx.x >> 5;
    const int lane = threadIdx.x & 31;
    const int b = blockIdx.x * 8 + wave;
    if (b >= Bn) return;

    const int start = lower_bound(index, NNZn, b);
    const int end   = lower_bound(index, NNZn, b + 1);

    v2f acc[En];
#pragma unroll
    for (int e = 0; e < En; ++e) { acc[e].x = 0.f; acc[e].y = 0.f; }
    float gacc = 0.f;
    const int gt = lane >> 3;   // task (lanes 0..15)
    const int ge = lane & 7;    // expert slot

    for (int i = start; i < end; ++i) {
        const int f = feats[i];
        const float v = values[i];
#pragma unroll
        for (int e = 0; e < En; ++e) {
            const float* p = w_exp0 + ((size_t)e * Vv + f) * Hn;
            v2f a = *(const v2f*)(p + 2 * lane);
            acc[e].x += a.x * v;
            acc[e].y += a.y * v;
        }
        if (lane < 16) {
            gacc += w_gates[((size_t)gt * Vv + f) * En + ge] * v;
        }
    }

#pragma unroll
    for (int e = 0; e < En; ++e) {
        v2f bb = *(const v2f*)(b_exp0 + e * Hn + 2 * lane);
        v2f h;
        h.x = fmaxf(acc[e].x + bb.x, 0.f);
        h.y = fmaxf(acc[e].y + bb.y, 0.f);
        *(v2f*)(h_out + ((size_t)e * Bn + b) * Hn + 2 * lane) = h;
    }
    if (lane < 16) logits[((size_t)gt * Bn + b) * En + ge] = gacc;
}

// Kernel 2: expert_out[e] = h[e] @ w_exp1[e] + b_exp1[e], via V_WMMA_F32_16X16X4_F32.
// One wave per (expert, 16-row B tile, 16-col O tile); K-loop over H=64 in steps of 4.
__global__ void mmoe_expert_wmma_kernel(const float* __restrict__ h,      // (E,B,H)
                                        const float* __restrict__ w_exp1, // (E,H,O)
                                        const float* __restrict__ b_exp1, // (E,O)
                                        float* __restrict__ eout)         // (E,B,O)
{
    const int wave = threadIdx.x >> 5;
    const int lane = threadIdx.x & 31;
    const int tile = blockIdx.x * 8 + wave;        // E * (B/16) * (O/16) tiles
    const int ntile = tile & 1;
    const int btile = (tile >> 1) & ((Bn / 16) - 1);
    const int e     = tile >> 11;

    const int m    = lane & 15;            // M (A) / N (B,C,D) position
    const int koff = (lane >> 4) << 1;     // 0 for lanes 0..15, 2 for 16..31
    const int b0   = btile * 16;
    const int n0   = ntile * 16;

    v8f acc = {};
    const float* Abase = h + ((size_t)e * Bn + b0 + m) * Hn;
    const float* Bbase = w_exp1 + (size_t)e * Hn * On;

#pragma unroll
    for (int k0 = 0; k0 < Hn; k0 += 4) {
        // A 16x4 f32 fragment: VGPR0 = A[m][k0+koff], VGPR1 = A[m][k0+koff+1]
        v2f a = *(const v2f*)(Abase + k0 + koff);
        // B 4x16 f32 fragment (mirror layout): N=m, rows koff / koff+1
        v2f bf;
        bf.x = Bbase[(size_t)(k0 + koff) * On + n0 + m];
        bf.y = Bbase[(size_t)(k0 + koff + 1) * On + n0 + m];
        acc = __builtin_amdgcn_wmma_f32_16x16x4_f32(
            /*neg_a=*/false, a, /*neg_b=*/false, bf,
            /*c_mod=*/(short)0, acc, /*reuse_a=*/false, /*reuse_b=*/false);
    }

    const float bias = b_exp1[e * On + n0 + m];
    const int mhi = (lane >> 4) * 8;       // C/D layout: VGPR r -> M = r (+8 for hi lanes)
#pragma unroll
    for (int r = 0; r < 8; ++r) {
        const int M = r + mhi;
        eout[((size_t)e * Bn + b0 + M) * On + n0 + m] = acc[r] + bias;
    }
}

// Kernel 3: per-row gating softmax + mixture + two towers. One wave per row.
__global__ void mmoe_tower_kernel(const float* __restrict__ logits, // (T,B,E)
                                  const float* __restrict__ eout,   // (E,B,O)
                                  const float* __restrict__ w_tow0, // (T,O,TH)
                                  const float* __restrict__ b_tow0, // (T,TH)
                                  const float* __restrict__ w_tow1, // (T,TH)
                                  const float* __restrict__ b_tow1, // (T)
                                  float* __restrict__ out)          // (B,T,1)
{
    const int wave = threadIdx.x >> 5;
    const int lane = threadIdx.x & 31;
    const int b = blockIdx.x * 8 + wave;
    if (b >= Bn) return;

#pragma unroll
    for (int t = 0; t < Tn; ++t) {
        // softmax over 8 logits (computed redundantly on every lane — broadcast loads)
        float l[En];
        float mx = -3.0e38f;
#pragma unroll
        for (int e = 0; e < En; ++e) {
            l[e] = logits[((size_t)t * Bn + b) * En + e];
            mx = fmaxf(mx, l[e]);
        }
        float s = 0.f;
#pragma unroll
        for (int e = 0; e < En; ++e) { l[e] = __expf(l[e] - mx); s += l[e]; }
        const float inv = 1.f / s;

        // mixture: ti[o], o = lane (O = 32)
        float ti = 0.f;
#pragma unroll
        for (int e = 0; e < En; ++e)
            ti += (l[e] * inv) * eout[((size_t)e * Bn + b) * On + lane];

        // tower layer 0: h[th] = relu(sum_o ti[o] * w_tow0[t][o][th] + b), th = lane
        float hacc = b_tow0[t * THn + lane];
#pragma unroll
        for (int o = 0; o < On; ++o)
            hacc += __shfl(ti, o, 32) * w_tow0[((size_t)t * On + o) * THn + lane];
        hacc = fmaxf(hacc, 0.f);

        // tower layer 1: scalar = sum_th h[th] * w_tow1[t][th]
        float dot = hacc * w_tow1[t * THn + lane];
#pragma unroll
        for (int off = 16; off > 0; off >>= 1)
            dot += __shfl_down(dot, off, 32);

        if (lane == 0) {
            const float z = dot + b_tow1[t];
            out[(size_t)b * Tn + t] = 1.f / (1.f + __expf(-z));
        }
    }
}

extern "C" void kernel_launch(void* const* d_in, const int* in_sizes, int n_in,
                              void* d_out, int out_size, void* d_ws, size_t ws_size,
                              hipStream_t stream) {
    const int*   index   = (const int*)d_in[1];
    const int*   feats   = (const int*)d_in[2];
    const float* values  = (const float*)d_in[3];
    const float* w_exp0  = (const float*)d_in[4];
    const float* b_exp0  = (const float*)d_in[5];
    const float* w_exp1  = (const float*)d_in[6];
    const float* b_exp1  = (const float*)d_in[7];
    const float* w_gates = (const float*)d_in[8];
    const float* w_tow0  = (const float*)d_in[9];
    const float* b_tow0  = (const float*)d_in[10];
    const float* w_tow1  = (const float*)d_in[11];
    const float* b_tow1  = (const float*)d_in[12];
    float* out = (float*)d_out;

    // workspace layout (f32): h (E,B,H) | logits (T,B,E) | expert_out (E,B,O)
    float* h_buf  = (float*)d_ws;
    float* logits = h_buf + (size_t)En * Bn * Hn;
    float* eout   = logits + (size_t)Tn * Bn * En;

    dim3 blk(256);
    mmoe_pool_kernel<<<Bn / 8, blk, 0, stream>>>(index, feats, values, w_exp0, b_exp0,
                                                 w_gates, h_buf, logits);
    const int tiles = En * (Bn / 16) * (On / 16);   // 16384 waves
    mmoe_expert_wmma_kernel<<<tiles / 8, blk, 0, stream>>>(h_buf, w_exp1, b_exp1, eout);
    mmoe_tower_kernel<<<Bn / 8, blk, 0, stream>>>(logits, eout, w_tow0, b_tow0,
                                                  w_tow1, b_tow1, out);
}